// VRFC_68015102099917
// MI455X (gfx1250) — compile-verified
//
#include <hip/hip_runtime.h>
#include <hip/hip_bf16.h>

typedef __attribute__((ext_vector_type(2))) float v2f;
typedef __attribute__((ext_vector_type(8))) float v8f;
typedef __attribute__((ext_vector_type(4))) int   v4i;

typedef __attribute__((address_space(1))) v4i glb_v4i;
typedef __attribute__((address_space(3))) v4i lds_v4i;

#define K_DIM   4096      // shared inner dim of every GEMM here
#define KC      64        // K-chunk staged in LDS
#define NCHUNK  (K_DIM / KC)
#define A_PITCH 68        // 32 rows x KC floats, pitch 68 -> 4-bank rotation per row
#define B_PITCH 160       // (KC/2) k-pairs x 64 cols x 2, pitch 160 -> 32-bank rotation
#define LDS_A_SZ (32 * A_PITCH)          // floats, one A buffer
#define LDS_B_SZ ((KC / 2) * B_PITCH)    // floats, one B buffer
#define LDS_TOTAL (2 * LDS_A_SZ + 2 * LDS_B_SZ)   // double buffered

// ---------------------------------------------------------------------------
// CDNA5 async global->LDS copy (ASYNCcnt-tracked), 16 bytes per lane.
// Builtin signature (from hipcc diagnostic): (v4i AS1*, v4i AS3*, int, int).
// ---------------------------------------------------------------------------
__device__ __forceinline__ void async_copy_b128(const float* g, float* l)
{
#if __has_builtin(__builtin_amdgcn_global_load_async_to_lds_b128)
    glb_v4i* gp = (glb_v4i*)(unsigned long long)(size_t)g;
    lds_v4i* lp = (lds_v4i*)(unsigned)(size_t)l;   // low 32 bits == LDS offset (flat aperture)
    __builtin_amdgcn_global_load_async_to_lds_b128(gp, lp, 0, 0);
#else
    unsigned lds_off = (unsigned)(size_t)l;
    asm volatile("global_load_async_to_lds_b128 %0, %1, off"
                 :: "v"(lds_off), "v"(g) : "memory");
#endif
}

__device__ __forceinline__ void wait_async0()
{
#if __has_builtin(__builtin_amdgcn_s_wait_asynccnt)
    __builtin_amdgcn_s_wait_asynccnt(0);
#else
    asm volatile("s_wait_asynccnt 0x0" ::: "memory");
#endif
}

// ---------------------------------------------------------------------------
// Core: one block computes C_tile[32 x 64] = A[row_base:+32, :4096] @ B[:4096, 0:51 pad 64]
// 8 waves; wave w owns the 16x16 sub-tile (m_tile = w>>2, n_tile = w&3).
// Double-buffered LDS: A filled by GLOBAL_LOAD_ASYNC_TO_LDS_B128 (no VGPR staging),
// B (tiny, needs pad/pack transform) staged via VGPRs; both overlap the WMMA phase.
// ---------------------------------------------------------------------------
__device__ __forceinline__ v8f tile_mm(const float* __restrict__ A, int M, int row_base,
                                       const float* __restrict__ B, int ldb,
                                       float* lds)
{
    const int t      = threadIdx.x;
    const int wave   = t >> 5;
    const int lane   = t & 31;
    const int half   = lane >> 4;
    const int l16    = lane & 15;
    const int m_tile = wave >> 2;    // 0..1
    const int n_tile = wave & 3;     // 0..3

    // --- A async-copy role: 32 rows x 64 cols, 2 x b128 per thread ---
    const int ar = t >> 3;           // 0..31
    const int ac = (t & 7) * 8;      // 0..56
    int arow = row_base + ar;
    if (arow >= M) arow = M - 1;     // clamp: uniform EXEC, dup rows are discarded at store
    const float* agp = A + (size_t)arow * K_DIM + ac;

    // --- B loader role: 64 k-rows x 64 n-cols, 16 scalars per thread ---
    const int bk = t >> 2;           // 0..63
    const int bc = (t & 3) * 16;     // 0,16,32,48
    const float* bgp = B + (size_t)bk * ldb;

    float* lA0 = lds;                               // A buffers
    float* lB0 = lds + 2 * LDS_A_SZ;                // B buffers

    float breg[16];
    v8f acc = {};

    // ---------------- prologue: fill buffer 0 with chunk 0 ----------------
    {
        float* dA = lA0 + ar * A_PITCH + ac;
        async_copy_b128(agp, dA);
        async_copy_b128(agp + 4, dA + 4);
        #pragma unroll
        for (int j = 0; j < 16; ++j) {
            const int n = bc + j;
            breg[j] = (n < 51) ? bgp[n] : 0.0f;
        }
        float* dB = lB0 + (bk >> 1) * B_PITCH + (bk & 1);
        #pragma unroll
        for (int j = 0; j < 16; ++j) dB[(bc + j) * 2] = breg[j];
    }
    wait_async0();
    __syncthreads();

    // ---------------- main ping-pong loop ----------------
    for (int i = 0; i < NCHUNK; ++i) {
        const int  cur      = i & 1;
        const int  nxt      = cur ^ 1;
        const bool has_next = (i + 1 < NCHUNK);
        const int  kb_next  = (i + 1) * KC;

        if (has_next) {
            // start chunk i+1: A straight into LDS (async), B into VGPRs
            float* dA = lA0 + nxt * LDS_A_SZ + ar * A_PITCH + ac;
            async_copy_b128(agp + kb_next, dA);
            async_copy_b128(agp + kb_next + 4, dA + 4);
            const float* bsrc = bgp + (size_t)kb_next * ldb;
            #pragma unroll
            for (int j = 0; j < 16; ++j) {
                const int n = bc + j;
                breg[j] = (n < 51) ? bsrc[n] : 0.0f;
            }
        }

        // --- 16 WMMA steps on the current buffers (overlaps the fills above) ---
        const float* aptr = lA0 + cur * LDS_A_SZ + (m_tile * 16 + l16) * A_PITCH + 2 * half;
        const float* bptr = lB0 + cur * LDS_B_SZ + half * B_PITCH + (n_tile * 16 + l16) * 2;
        #pragma unroll
        for (int k0 = 0; k0 < KC; k0 += 4) {
            // A frag: a.x=A[m][k0+2h], a.y=A[m][k0+2h+1]   (ISA 16x4 f32 layout)
            v2f a = *(const v2f*)(aptr + k0);
            // B frag: b.x=B[k0+2h][n], b.y=B[k0+2h+1][n]   (packed pair -> ds_load_b64)
            v2f b = *(const v2f*)(bptr + (k0 >> 1) * B_PITCH);
            acc = __builtin_amdgcn_wmma_f32_16x16x4_f32(false, a, false, b,
                                                        (short)0, acc, false, false);
        }

        if (has_next) {
            // land B into the back buffer (loads had the whole compute phase to arrive)
            float* dB = lB0 + nxt * LDS_B_SZ + (bk >> 1) * B_PITCH + (bk & 1);
            #pragma unroll
            for (int j = 0; j < 16; ++j) dB[(bc + j) * 2] = breg[j];
            wait_async0();           // back-buffer A complete
        }
        __syncthreads();
    }
    return acc;
}

// ---------------------------------------------------------------------------
// Generic skinny GEMM: C[M x 64(ld)] = A[M x 4096] @ B[:4096, 0:51] (cols 51..63 -> 0)
// ---------------------------------------------------------------------------
__global__ void gemm51_kernel(const float* __restrict__ A, int M,
                              const float* __restrict__ B, int ldb,
                              float* __restrict__ C)
{
    __shared__ float lds[LDS_TOTAL];
    const int row_base = blockIdx.x * 32;
    v8f acc = tile_mm(A, M, row_base, B, ldb, lds);

    const int t = threadIdx.x;
    const int wave = t >> 5, lane = t & 31, half = lane >> 4, l16 = lane & 15;
    const int m_tile = wave >> 2, n_tile = wave & 3;
    const int n = n_tile * 16 + l16;
    #pragma unroll
    for (int v = 0; v < 8; ++v) {
        const int row = row_base + m_tile * 16 + half * 8 + v;  // C/D layout: M = v + 8*half
        if (row < M) C[(size_t)row * 64 + n] = acc[v];
    }
}

// ---------------------------------------------------------------------------
// Fused final GEMM + gather epilogue:
// rel_dists[r,n] = (vr @ Wp3)[r,n] + P1[sub[r],n] + P2[obj[r],n] + G4[img[r],n] + bias[n]
// ---------------------------------------------------------------------------
__global__ void final_kernel(const float* __restrict__ vr,  const float* __restrict__ Wp3,
                             const float* __restrict__ P1,  const float* __restrict__ P2,
                             const float* __restrict__ G4,  const float* __restrict__ bp3,
                             const int*   __restrict__ rel_inds,
                             float* __restrict__ out)
{
    __shared__ float lds[LDS_TOTAL];
    const int row_base = blockIdx.x * 32;
    v8f acc = tile_mm(vr, 20000, row_base, Wp3, 64, lds);

    const int t = threadIdx.x;
    const int wave = t >> 5, lane = t & 31, half = lane >> 4, l16 = lane & 15;
    const int m_tile = wave >> 2, n_tile = wave & 3;
    const int n = n_tile * 16 + l16;
    if (n < 51) {
        const float bn = bp3[n];
        #pragma unroll
        for (int v = 0; v < 8; ++v) {
            const int row = row_base + m_tile * 16 + half * 8 + v;
            const int img = rel_inds[row * 3 + 0];
            const int sub = rel_inds[row * 3 + 1];
            const int obj = rel_inds[row * 3 + 2];
            const float val = acc[v] + bn
                            + P1[(size_t)sub * 64 + n]
                            + P2[(size_t)obj * 64 + n]
                            + G4[(size_t)img * 64 + n];
            out[(size_t)row * 51 + n] = val;
        }
    }
}

// --------------------------- small helper kernels ---------------------------
__global__ void copy_f4_kernel(const float4* __restrict__ src, float4* __restrict__ dst, int n4)
{
    const int i = blockIdx.x * blockDim.x + threadIdx.x;
    if (i < n4) dst[i] = src[i];
}

__global__ void argmax_kernel(const float* __restrict__ logits, float* __restrict__ pred, int rows)
{
    const int r = blockIdx.x * blockDim.x + threadIdx.x;
    if (r >= rows) return;
    const float* p = logits + (size_t)r * 151;
    float best = p[1];
    int   bi   = 1;                         // argmax over [:,1:] + 1 == original column
    for (int c = 2; c < 151; ++c) {
        const float v = p[c];
        if (v > best) { best = v; bi = c; } // strict > keeps first-index tie rule
    }
    pred[r] = (float)bi;
}

// bias[n] = b_fc[n] + sum_k b_proj[k] * W_fc[8192+k, n]   (n padded to 64)
__global__ void bias_kernel(const float* __restrict__ b_proj, const float* __restrict__ W_fc,
                            const float* __restrict__ b_fc,   float* __restrict__ bp3)
{
    __shared__ float partial[4][64];
    const int n    = threadIdx.x & 63;
    const int part = threadIdx.x >> 6;      // 4-way K split
    float s = 0.0f;
    if (n < 51) {
        for (int k = part * 1024; k < (part + 1) * 1024; ++k)
            s += b_proj[k] * W_fc[(size_t)(8192 + k) * 51 + n];
    }
    partial[part][n] = s;
    __syncthreads();
    if (part == 0) {
        const float tot = partial[0][n] + partial[1][n] + partial[2][n] + partial[3][n];
        bp3[n] = ((n < 51) ? b_fc[n] : 0.0f) + tot;
    }
}

// ---------------------------------------------------------------------------
extern "C" void kernel_launch(void* const* d_in, const int* in_sizes, int n_in,
                              void* d_out, int out_size, void* d_ws, size_t ws_size,
                              hipStream_t stream)
{
    const float* obj_fmaps       = (const float*)d_in[0];  // 5000 x 4096
    const float* obj_logits      = (const float*)d_in[1];  // 5000 x 151
    const float* vr              = (const float*)d_in[2];  // 20000 x 4096
    const int*   rel_inds        = (const int*)  d_in[3];  // 20000 x 3 (img,sub,obj)
    const float* global_features = (const float*)d_in[4];  // 64 x 4096
    const float* W_proj          = (const float*)d_in[5];  // 4096 x 4096
    const float* b_proj          = (const float*)d_in[6];  // 4096
    const float* W_fc            = (const float*)d_in[7];  // 16384 x 51
    const float* b_fc            = (const float*)d_in[8];  // 51

    float* out = (float*)d_out;
    // d_out layout: obj_dists2 [755000] | obj_preds [5000] | rel_dists [1020000]
    float* out_logits = out;
    float* out_preds  = out + 755000;
    float* out_rel    = out + 760000;

    // workspace layout (floats): P1[5000x64] P2[5000x64] Wp3[4096x64] G4[64x64] bias[64]
    float* ws  = (float*)d_ws;
    float* P1  = ws;
    float* P2  = ws + 320000;
    float* Wp3 = ws + 640000;
    float* G4  = ws + 902144;
    float* bp3 = ws + 906240;

    const float* W1 = W_fc;                        // rows 0..4095     (obj_fmaps[sub])
    const float* W2 = W_fc + (size_t)4096  * 51;   // rows 4096..8191  (obj_fmaps[obj])
    const float* W3 = W_fc + (size_t)8192  * 51;   // rows 8192..12287 (vr_p path)
    const float* W4 = W_fc + (size_t)12288 * 51;   // rows 12288..     (global_features)

    // outputs 1 & 2
    copy_f4_kernel<<<(188750 + 255) / 256, 256, 0, stream>>>(
        (const float4*)obj_logits, (float4*)out_logits, 188750);
    argmax_kernel<<<(5000 + 255) / 256, 256, 0, stream>>>(obj_logits, out_preds, 5000);

    // bias[n] = b_fc[n] + b_proj @ W3
    bias_kernel<<<1, 256, 0, stream>>>(b_proj, W_fc, b_fc, bp3);

    // precomputed skinny GEMMs (all K=4096, N=51 padded to 64)
    gemm51_kernel<<<157, 256, 0, stream>>>(obj_fmaps, 5000, W1, 51, P1);
    gemm51_kernel<<<157, 256, 0, stream>>>(obj_fmaps, 5000, W2, 51, P2);
    gemm51_kernel<<<128, 256, 0, stream>>>(W_proj, 4096, W3, 51, Wp3);   // W_proj @ W3
    gemm51_kernel<<<2,   256, 0, stream>>>(global_features, 64, W4, 51, G4);

    // rel_dists = vr @ Wp3 + gathers + bias   (20000 rows = 625 blocks of 32)
    final_kernel<<<625, 256, 0, stream>>>(vr, Wp3, P1, P2, G4, bp3, rel_inds, out_rel);
}